// GCNNet_2156073583056
// MI455X (gfx1250) — compile-verified
//
#include <hip/hip_runtime.h>
#include <cstdint>

#define NEG_SLOPE 0.2f

typedef float v2f __attribute__((ext_vector_type(2)));
typedef float v8f __attribute__((ext_vector_type(8)));

// ---------------------------------------------------------------------------
// init: m = -FLT_MAX, s = 0 (first n), acc = 0 (first accN)
// ---------------------------------------------------------------------------
__global__ void k_init(float* __restrict__ m, float* __restrict__ s,
                       float* __restrict__ acc, int n, long long accN) {
  long long i = (long long)blockIdx.x * blockDim.x + threadIdx.x;
  if (i < n) { m[i] = -3.402823466e38f; s[i] = 0.0f; }
  if (i < accN) acc[i] = 0.0f;
}

// ---------------------------------------------------------------------------
// H[n, Cout] = X[n, Cin] @ W[Cin, Cout] using fp32 WMMA 16x16x4.
// One wave32 per 16x16 output tile. n % 16 == 0, Cout % 16 == 0 (guaranteed).
// A layout (16x4 f32): lanes 0-15 rows, v0={K0|K2}, v1={K1|K3} (halves).
// B layout (4x16 f32): lanes 0-15 cols, v0={K0|K2}, v1={K1|K3}.
// C/D layout: VGPR v -> row v (+8 for upper half), col = lane%16.
// ---------------------------------------------------------------------------
__global__ void k_gemm_wmma(const float* __restrict__ X, const float* __restrict__ W,
                            float* __restrict__ H, int n, int Cin, int Cout) {
  int wave = (int)((blockIdx.x * blockDim.x + threadIdx.x) >> 5);
  int lane = threadIdx.x & 31;
  int ntn = Cout >> 4;
  int total = (n >> 4) * ntn;
  if (wave >= total) return;               // whole wave exits together (EXEC stays full)
  int mt = wave / ntn, nt = wave - mt * ntn;
  int row0 = mt << 4, col0 = nt << 4;
  int r = lane & 15, half = lane >> 4;

  const float* __restrict__ xrow = X + (size_t)(row0 + r) * Cin;
  const float* __restrict__ wcol = W + col0 + r;

  v8f acc = {};
  for (int k = 0; k < Cin; k += 4) {
    int ka = k + 2 * half;
    v2f a, b;
    a.x = xrow[ka];
    a.y = xrow[ka + 1];
    b.x = wcol[(size_t)ka * Cout];
    b.y = wcol[(size_t)(ka + 1) * Cout];
    acc = __builtin_amdgcn_wmma_f32_16x16x4_f32(false, a, false, b,
                                                (short)0, acc, false, false);
  }
#pragma unroll
  for (int v = 0; v < 8; ++v) {
    H[(size_t)(row0 + v + 8 * half) * Cout + (col0 + r)] = acc[v];
  }
}

// ---------------------------------------------------------------------------
// e_src[i] = h[i,:] . a_src ; e_dst[i] = h[i,:] . a_dst  (one wave32 per node)
// ---------------------------------------------------------------------------
__global__ void k_attn_proj(const float* __restrict__ H,
                            const float* __restrict__ a_s, const float* __restrict__ a_d,
                            float* __restrict__ es, float* __restrict__ ed,
                            int n, int Cout) {
  int wave = (int)((blockIdx.x * blockDim.x + threadIdx.x) >> 5);
  int lane = threadIdx.x & 31;
  if (wave >= n) return;
  const float* __restrict__ hr = H + (size_t)wave * Cout;
  float s0 = 0.f, s1 = 0.f;
  for (int c = lane; c < Cout; c += 32) {
    float h = hr[c];
    s0 += h * a_s[c];
    s1 += h * a_d[c];
  }
#pragma unroll
  for (int off = 16; off > 0; off >>= 1) {
    s0 += __shfl_down(s0, off, 32);
    s1 += __shfl_down(s1, off, 32);
  }
  if (lane == 0) { es[wave] = s0; ed[wave] = s1; }
}

// ordered-int trick: correct float atomic max for any sign mix,
// provided init is -FLT_MAX (which encodes consistently).
__device__ __forceinline__ void atomicMaxF32(float* addr, float v) {
  if (v >= 0.0f) atomicMax((int*)addr, __float_as_int(v));
  else           atomicMin((unsigned int*)addr, __float_as_uint(v));
}

// ---------------------------------------------------------------------------
// le[e] = leaky_relu(es[src] + ed[dst]); m[dst] = max(m[dst], le[e])
// Edges [0,E) from edge_index; edges [E, E+n) are self loops.
// ---------------------------------------------------------------------------
__global__ void k_edge_logit_max(const int* __restrict__ ei, int E, int n,
                                 const float* __restrict__ es, const float* __restrict__ ed,
                                 float* __restrict__ le, float* __restrict__ m) {
  int idx = blockIdx.x * blockDim.x + threadIdx.x;
  int Et = E + n;
  if (idx >= Et) return;
  int s_, d_;
  if (idx < E) { s_ = ei[idx]; d_ = ei[E + idx]; }
  else         { s_ = d_ = idx - E; }
  float e = es[s_] + ed[d_];
  e = (e > 0.0f) ? e : NEG_SLOPE * e;
  le[idx] = e;
  atomicMaxF32(&m[d_], e);
}

// ---------------------------------------------------------------------------
// le[e] = exp(le[e] - m[dst]); s[dst] += le[e]
// ---------------------------------------------------------------------------
__global__ void k_edge_exp_sum(const int* __restrict__ ei, int E, int n,
                               float* __restrict__ le, const float* __restrict__ m,
                               float* __restrict__ s) {
  int idx = blockIdx.x * blockDim.x + threadIdx.x;
  int Et = E + n;
  if (idx >= Et) return;
  int d_ = (idx < E) ? ei[E + idx] : (idx - E);
  float ex = __expf(le[idx] - m[d_]);
  le[idx] = ex;
  atomicAdd(&s[d_], ex);
}

// ---------------------------------------------------------------------------
// acc[dst, c] += (le[e]/s[dst]) * h[src, c]
// thread = flat (edge, channel); consecutive threads hit consecutive channels
// -> coalesced loads and coalesced atomic lanes into L2.
// ---------------------------------------------------------------------------
__global__ void k_edge_aggregate(const int* __restrict__ ei, int E, int n, int Cout,
                                 const float* __restrict__ le, const float* __restrict__ s,
                                 const float* __restrict__ H, float* __restrict__ acc) {
  long long t = (long long)blockIdx.x * blockDim.x + threadIdx.x;
  long long total = (long long)(E + n) * Cout;
  if (t >= total) return;
  int edge = (int)(t / Cout);
  int c    = (int)(t - (long long)edge * Cout);
  int s_, d_;
  if (edge < E) { s_ = ei[edge]; d_ = ei[E + edge]; }
  else          { s_ = d_ = edge - E; }
  float alpha = le[edge] / s[d_];
  atomicAdd(&acc[(size_t)d_ * Cout + c], alpha * H[(size_t)s_ * Cout + c]);
}

// ---------------------------------------------------------------------------
// act = relu(acc + b)
// ---------------------------------------------------------------------------
__global__ void k_bias_relu(const float* __restrict__ acc, const float* __restrict__ b,
                            float* __restrict__ act, long long total, int Cout) {
  long long t = (long long)blockIdx.x * blockDim.x + threadIdx.x;
  if (t >= total) return;
  int c = (int)(t % Cout);
  float v = acc[t] + b[c];
  act[t] = v > 0.0f ? v : 0.0f;
}

// ---------------------------------------------------------------------------
// out[i,o] = act[i,:] . fcW[:,o] + fcb[o]   (Co = 40, scalar path)
// ---------------------------------------------------------------------------
__global__ void k_fc(const float* __restrict__ A, const float* __restrict__ W,
                     const float* __restrict__ b, float* __restrict__ out,
                     int n, int Cin, int Co) {
  long long t = (long long)blockIdx.x * blockDim.x + threadIdx.x;
  if (t >= (long long)n * Co) return;
  int i = (int)(t / Co);
  int o = (int)(t - (long long)i * Co);
  const float* __restrict__ ar = A + (size_t)i * Cin;
  float acc = b[o];
  for (int c = 0; c < Cin; ++c) acc += ar[c] * W[(size_t)c * Co + o];
  out[t] = acc;
}

// ---------------------------------------------------------------------------
extern "C" void kernel_launch(void* const* d_in, const int* in_sizes, int n_in,
                              void* d_out, int out_size, void* d_ws, size_t ws_size,
                              hipStream_t stream) {
  const float* x  = (const float*)d_in[0];
  const int*   ei = (const int*)d_in[1];
  const int dims[5] = {256, 32, 64, 128, 128};
  const int n  = in_sizes[0] / dims[0];   // 50000
  const int E  = in_sizes[1] / 2;         // 800000
  const int Et = E + n;

  // workspace layout (floats)
  float* ws     = (float*)d_ws;
  size_t big    = (size_t)n * 128;
  float* bufH   = ws;                 // h of current layer
  float* bufACC = bufH + big;         // aggregation accumulator
  float* bufA   = bufACC + big;       // activation ping buffer (layer output)
  float* es     = bufA + big;
  float* ed     = es + n;
  float* mbuf   = ed + n;
  float* sbuf   = mbuf + n;
  float* le     = sbuf + n;           // Et floats

  const int THR = 256;
  const float* act_in = x;

  for (int l = 0; l < 4; ++l) {
    int Cin = dims[l], Cout = dims[l + 1];
    const float* W  = (const float*)d_in[2 + l * 4 + 0];
    const float* as = (const float*)d_in[2 + l * 4 + 1];
    const float* ad = (const float*)d_in[2 + l * 4 + 2];
    const float* b  = (const float*)d_in[2 + l * 4 + 3];

    long long accN = (long long)n * Cout;

    k_init<<<(unsigned)((accN + THR - 1) / THR), THR, 0, stream>>>(mbuf, sbuf, bufACC, n, accN);

    int tiles   = (n / 16) * (Cout / 16);
    long long gthreads = (long long)tiles * 32;
    k_gemm_wmma<<<(unsigned)((gthreads + THR - 1) / THR), THR, 0, stream>>>(
        act_in, W, bufH, n, Cin, Cout);

    long long pthreads = (long long)n * 32;
    k_attn_proj<<<(unsigned)((pthreads + THR - 1) / THR), THR, 0, stream>>>(
        bufH, as, ad, es, ed, n, Cout);

    k_edge_logit_max<<<(unsigned)((Et + THR - 1) / THR), THR, 0, stream>>>(
        ei, E, n, es, ed, le, mbuf);

    k_edge_exp_sum<<<(unsigned)((Et + THR - 1) / THR), THR, 0, stream>>>(
        ei, E, n, le, mbuf, sbuf);

    long long athreads = (long long)Et * Cout;
    k_edge_aggregate<<<(unsigned)((athreads + THR - 1) / THR), THR, 0, stream>>>(
        ei, E, n, Cout, le, sbuf, bufH, bufACC);

    k_bias_relu<<<(unsigned)((accN + THR - 1) / THR), THR, 0, stream>>>(
        bufACC, b, bufA, accN, Cout);

    act_in = bufA;
  }

  const float* fcW = (const float*)d_in[18];
  const float* fcb = (const float*)d_in[19];
  long long othreads = (long long)n * 40;
  k_fc<<<(unsigned)((othreads + THR - 1) / THR), THR, 0, stream>>>(
      act_in, fcW, fcb, (float*)d_out, n, 128, 40);
}